// HyperbolicGraphConvolution_9088150798638
// MI455X (gfx1250) — compile-verified
//
#include <hip/hip_runtime.h>
#include <hip/hip_bf16.h>

typedef __attribute__((ext_vector_type(2))) float v2f;
typedef __attribute__((ext_vector_type(8))) float v8f;

#define N_NODES 8192
#define DIM 256
#define MAXNORM 0.996f   // (1 - 4e-3)/sqrt(c), c = 1
#define MIN_NORM 1e-15f

// ---------------------------------------------------------------------------
// helpers
// ---------------------------------------------------------------------------
__device__ __forceinline__ float artanh_c(float x) {
    x = fminf(fmaxf(x, -1.f + 1e-7f), 1.f - 1e-7f);
    return 0.5f * logf((1.f + x) / (1.f - x));
}

// block-wide sum over 256 threads (8 wave32s); returns the same value to all
__device__ __forceinline__ float blockReduceSum(float v, float* sm) {
    #pragma unroll
    for (int off = 16; off > 0; off >>= 1)
        v += __shfl_xor(v, off, 32);
    const int wid = threadIdx.x >> 5;
    __syncthreads();                      // protect sm reuse across calls
    if ((threadIdx.x & 31) == 0) sm[wid] = v;
    __syncthreads();
    float r = sm[0];
    #pragma unroll
    for (int i = 1; i < 8; ++i) r += sm[i];
    return r;
}

// ---------------------------------------------------------------------------
// Kernel 1: mx = x @ W^T   (M=8192, K=256, N=256), f32 WMMA 16x16x4
// wave tile: 16 rows x 64 cols (4 accum tiles). 2048 waves = 256 blocks.
// ---------------------------------------------------------------------------
__global__ __launch_bounds__(256)
void wmma_gemm_xwT(const float* __restrict__ x, const float* __restrict__ w,
                   float* __restrict__ mx) {
    const int wid  = threadIdx.x >> 5;
    const int lane = threadIdx.x & 31;
    const int gw   = blockIdx.x * 8 + wid;
    const int r0   = (gw >> 2) * 16;       // 512 row strips
    const int c0   = (gw & 3) * 64;        // 4 column groups
    const int lr   = lane & 15;
    const int lk2  = (lane >> 4) * 2;      // 0 or 2

    v8f acc[4] = {};
    const float* arow = x + (size_t)(r0 + lr) * DIM + lk2;
    for (int k = 0; k < DIM; k += 4) {
        v2f a = *(const v2f*)(arow + k);
        #pragma unroll
        for (int t = 0; t < 4; ++t) {
            const int col = c0 + t * 16 + lr;
            v2f b = *(const v2f*)(w + (size_t)col * DIM + k + lk2);
            acc[t] = __builtin_amdgcn_wmma_f32_16x16x4_f32(
                false, a, false, b, (short)0, acc[t], false, false);
        }
    }
    const int rbase = r0 + (lane >> 4) * 8;
    #pragma unroll
    for (int t = 0; t < 4; ++t) {
        const int col = c0 + t * 16 + lr;
        #pragma unroll
        for (int v = 0; v < 8; ++v)
            mx[(size_t)(rbase + v) * DIM + col] = acc[t][v];
    }
}

// ---------------------------------------------------------------------------
// Kernel 2: HypLinear epilogue + logmap0; writes x_tangent TRANSPOSED
// one block (256 threads) per row n
// ---------------------------------------------------------------------------
__global__ __launch_bounds__(256)
void hyp_linear_epilogue(const float* __restrict__ x, const float* __restrict__ mx,
                         const float* __restrict__ bias, float* __restrict__ xtT) {
    __shared__ float sm[8];
    const int n = blockIdx.x;
    const int d = threadIdx.x;
    const float xv = x[(size_t)n * DIM + d];
    const float mv = mx[(size_t)n * DIM + d];
    const float bv = bias[d];

    // mobius_matvec tail
    const float x2 = blockReduceSum(xv * xv, sm);
    const float m2 = blockReduceSum(mv * mv, sm);
    const float xn = fmaxf(sqrtf(x2), MIN_NORM);
    const float mn = fmaxf(sqrtf(m2), MIN_NORM);
    const float sc = tanhf(mn / xn * artanh_c(xn)) / mn;   // sqrt_c = 1
    float res = (m2 == 0.f) ? 0.f : mv * sc;
    // proj(res): ||res|| = sc * mn  (sc >= 0)
    float rn = (m2 == 0.f) ? MIN_NORM : fmaxf(sc * mn, MIN_NORM);
    if (rn > MAXNORM) res *= MAXNORM / rn;

    // hyp_bias = proj(expmap0(bias))
    const float b2 = blockReduceSum(bv * bv, sm);
    const float bn = fmaxf(sqrtf(b2), MIN_NORM);
    float hb = tanhf(bn) / bn * bv;
    const float hbn = fmaxf(tanhf(bn), MIN_NORM);
    if (hbn > MAXNORM) hb *= MAXNORM / hbn;

    // mobius_add(res, hb)
    const float x2a = blockReduceSum(res * res, sm);
    const float y2a = blockReduceSum(hb * hb, sm);
    const float xya = blockReduceSum(res * hb, sm);
    const float num = (1.f + 2.f * xya + y2a) * res + (1.f - x2a) * hb;
    const float den = fmaxf(1.f + 2.f * xya + x2a * y2a, MIN_NORM);
    float h = num / den;

    // proj
    const float h2 = blockReduceSum(h * h, sm);
    float hn = fmaxf(sqrtf(h2), MIN_NORM);
    if (hn > MAXNORM) { h *= MAXNORM / hn; hn = MAXNORM; }

    // logmap0 -> transposed store (column-contiguous B for the agg GEMM)
    const float s = artanh_c(hn) / hn;
    xtT[(size_t)d * N_NODES + n] = h * s;
}

// ---------------------------------------------------------------------------
// Kernel 3: support = adj @ x_tangent  (M=K=8192, N=256), f32 WMMA 16x16x4
// wave tile: 16 rows x 128 cols (8 accum tiles). 1024 waves = 128 blocks.
// Exact sparsity skip: all-zero 16x4 adj tiles contribute nothing.
// ---------------------------------------------------------------------------
__global__ __launch_bounds__(256)
void wmma_gemm_agg(const float* __restrict__ adj, const float* __restrict__ xtT,
                   float* __restrict__ sup) {
    const int wid  = threadIdx.x >> 5;
    const int lane = threadIdx.x & 31;
    const int gw   = blockIdx.x * 8 + wid;
    const int r0   = (gw >> 1) * 16;       // 512 row strips
    const int c0   = (gw & 1) * 128;       // 2 column groups
    const int lr   = lane & 15;
    const int lk2  = (lane >> 4) * 2;

    v8f acc[8] = {};
    const float* arow = adj + (size_t)(r0 + lr) * N_NODES + lk2;
    for (int k = 0; k < N_NODES; k += 4) {
        v2f a = *(const v2f*)(arow + k);
        // uniform scalar branch: EXEC stays all-ones for WMMA legality
        if (__builtin_amdgcn_ballot_w32(a.x != 0.f || a.y != 0.f) == 0u)
            continue;
        #pragma unroll
        for (int t = 0; t < 8; ++t) {
            const int col = c0 + t * 16 + lr;
            v2f b = *(const v2f*)(xtT + (size_t)col * N_NODES + k + lk2);
            acc[t] = __builtin_amdgcn_wmma_f32_16x16x4_f32(
                false, a, false, b, (short)0, acc[t], false, false);
        }
    }
    const int rbase = r0 + (lane >> 4) * 8;
    #pragma unroll
    for (int t = 0; t < 8; ++t) {
        const int col = c0 + t * 16 + lr;
        #pragma unroll
        for (int v = 0; v < 8; ++v)
            sup[(size_t)(rbase + v) * DIM + col] = acc[t][v];
    }
}

// ---------------------------------------------------------------------------
// Kernel 4: h = proj(expmap0(sup)); xt = relu(logmap0(h)); out = proj(expmap0(xt))
// ---------------------------------------------------------------------------
__global__ __launch_bounds__(256)
void hyp_agg_act(const float* __restrict__ sup, float* __restrict__ out) {
    __shared__ float sm[8];
    const int n = blockIdx.x;
    const int d = threadIdx.x;
    const float u = sup[(size_t)n * DIM + d];

    const float u2 = blockReduceSum(u * u, sm);
    const float un = fmaxf(sqrtf(u2), MIN_NORM);
    float h = tanhf(un) / un * u;
    float hn = fmaxf(tanhf(un), MIN_NORM);
    if (hn > MAXNORM) { h *= MAXNORM / hn; hn = MAXNORM; }

    float xt = artanh_c(hn) / hn * h;
    xt = fmaxf(xt, 0.f);   // relu

    const float t2 = blockReduceSum(xt * xt, sm);
    const float tn = fmaxf(sqrtf(t2), MIN_NORM);
    float h2 = tanhf(tn) / tn * xt;
    const float h2n = fmaxf(tanhf(tn), MIN_NORM);
    if (h2n > MAXNORM) h2 *= MAXNORM / h2n;

    out[(size_t)n * DIM + d] = h2;
}

// ---------------------------------------------------------------------------
extern "C" void kernel_launch(void* const* d_in, const int* in_sizes, int n_in,
                              void* d_out, int out_size, void* d_ws, size_t ws_size,
                              hipStream_t stream) {
    const float* x    = (const float*)d_in[0];
    const float* adj  = (const float*)d_in[1];
    const float* w    = (const float*)d_in[2];
    const float* bias = (const float*)d_in[3];
    float* out = (float*)d_out;

    // workspace: mx (8 MB), xtT (8 MB); sup reuses mx's region (16 MB total)
    float* mx  = (float*)d_ws;
    float* xtT = mx + (size_t)N_NODES * DIM;
    float* sup = mx;

    wmma_gemm_xwT<<<dim3(256), dim3(256), 0, stream>>>(x, w, mx);
    hyp_linear_epilogue<<<dim3(N_NODES), dim3(DIM), 0, stream>>>(x, mx, bias, xtT);
    wmma_gemm_agg<<<dim3(128), dim3(256), 0, stream>>>(adj, xtT, sup);
    hyp_agg_act<<<dim3(N_NODES), dim3(DIM), 0, stream>>>(sup, out);

    // second output of the tuple: adj passthrough (graph-capture-safe D2D copy)
    hipMemcpyAsync(out + (size_t)N_NODES * DIM, adj,
                   (size_t)N_NODES * N_NODES * sizeof(float),
                   hipMemcpyDeviceToDevice, stream);
}